// NeuralSplineFourierFilter_28750511079899
// MI455X (gfx1250) — compile-verified
//
#include <hip/hip_runtime.h>

// ---------------------------------------------------------------------------
// NeuralSplineFourierFilter for MI455X (gfx1250, wave32)
//
// Kernel 1 (1 wave): MLP via V_WMMA_F32_16X16X4_F32, then collapse the cubic
//   B-spline into 7 per-interval power-basis cubics. Writes a 36-float table
//   to d_ws:  [0..5] = interior breakpoints kpos[1..6], [6..7] pad,
//             [8+4*i .. 8+4*i+3] = cubic coeffs (c0..c3) for interval i.
// Kernel 2 (streaming, memory-bound ~5.8us at 23.3 TB/s): per element
//   xp = clamp(x/sqrt(3)); idx = #(xp >= brk[i]); Horner with coef[idx].
// ---------------------------------------------------------------------------

typedef __attribute__((ext_vector_type(2))) float v2f;
typedef __attribute__((ext_vector_type(8))) float v8f;

#define LATENT 8   // (unused placeholder guard)
#undef LATENT

__global__ void __launch_bounds__(32)
mlp_setup_kernel(const float* __restrict__ a,
                 const float* __restrict__ W1, const float* __restrict__ b1,
                 const float* __restrict__ W2, const float* __restrict__ b2,
                 const float* __restrict__ Ww, const float* __restrict__ bw,
                 const float* __restrict__ Wk, const float* __restrict__ bk,
                 float* __restrict__ tbl)
{
    __shared__ float s_net[32];
    __shared__ float s_head[16];
    __shared__ float s_ak[14];
    __shared__ float s_w[10];

    const int L  = threadIdx.x;      // one wave32
    const int m  = L & 15;           // column / M index within half-wave
    const int hi = L >> 4;           // half-wave select

    // ---- layer 1: net1 = sin(a * W1 + b1), one neuron per lane ----
    const float av = a[0];
    s_net[L] = sinf(fmaf(av, W1[L], b1[L]));
    __syncthreads();

    // ---- layer 2: (1x32) @ (32x32) via WMMA f32 16x16x4 ----
    // A tile (16x4, only row M=0 nonzero): lane m, hi -> (M=m, K=2*hi+v)
    // B tile (4x16): VGPR v holds rows {v | v+2} across lane halves -> K=2*hi+v
    v8f acc0 = {0.f,0.f,0.f,0.f,0.f,0.f,0.f,0.f};
    v8f acc1 = {0.f,0.f,0.f,0.f,0.f,0.f,0.f,0.f};
#pragma unroll
    for (int kk = 0; kk < 8; ++kk) {
        const int k0 = 4 * kk + 2 * hi;
        v2f af;
        af.x = (m == 0) ? s_net[k0 + 0] : 0.f;
        af.y = (m == 0) ? s_net[k0 + 1] : 0.f;
        v2f bf0, bf1;
        bf0.x = W2[(k0 + 0) * 32 + m];
        bf0.y = W2[(k0 + 1) * 32 + m];
        bf1.x = W2[(k0 + 0) * 32 + 16 + m];
        bf1.y = W2[(k0 + 1) * 32 + 16 + m];
        acc0 = __builtin_amdgcn_wmma_f32_16x16x4_f32(false, af, false, bf0,
                                                     (short)0, acc0, false, false);
        acc1 = __builtin_amdgcn_wmma_f32_16x16x4_f32(false, af, false, bf1,
                                                     (short)0, acc1, false, false);
    }
    __syncthreads();
    // D row M=0 lives in VGPR0 of lanes 0..15 (C/D layout table)
    if (L < 16) {
        s_net[L]      = sinf(acc0[0] + b2[L]);
        s_net[L + 16] = sinf(acc1[0] + b2[L + 16]);
    }
    __syncthreads();

    // ---- heads: (1x32) @ (32x16) where cols = [Ww(9) | Wk(7)] ----
    v8f acch = {0.f,0.f,0.f,0.f,0.f,0.f,0.f,0.f};
#pragma unroll
    for (int kk = 0; kk < 8; ++kk) {
        const int k0 = 4 * kk + 2 * hi;
        v2f af;
        af.x = (m == 0) ? s_net[k0 + 0] : 0.f;
        af.y = (m == 0) ? s_net[k0 + 1] : 0.f;
        v2f bf;
        bf.x = (m < 9) ? Ww[(k0 + 0) * 9 + m] : Wk[(k0 + 0) * 7 + (m - 9)];
        bf.y = (m < 9) ? Ww[(k0 + 1) * 9 + m] : Wk[(k0 + 1) * 7 + (m - 9)];
        acch = __builtin_amdgcn_wmma_f32_16x16x4_f32(false, af, false, bf,
                                                     (short)0, acch, false, false);
    }
    if (L < 16) {
        s_head[L] = acch[0] + ((L < 9) ? bw[L] : bk[L - 9]);
    }
    __syncthreads();

    // ---- lane 0: softmax -> cumsum -> knot vector ak[14], ctrl pts w[10] ----
    if (L == 0) {
        float kr[7], e[7];
        float mx = s_head[9];
        for (int i = 1; i < 7; ++i) mx = fmaxf(mx, s_head[9 + i]);
        float ssum = 0.f;
        for (int i = 0; i < 7; ++i) { kr[i] = s_head[9 + i]; e[i] = expf(kr[i] - mx); ssum += e[i]; }
        const float inv = 1.0f / ssum;
        s_ak[0] = s_ak[1] = s_ak[2] = 0.f;
        s_ak[3] = 0.f;                       // kpos[0]
        float cum = 0.f;
        for (int i = 0; i < 7; ++i) { cum += e[i] * inv; s_ak[4 + i] = cum; }
        s_ak[11] = s_ak[12] = s_ak[13] = 1.f;
        s_w[0] = 0.f;
        for (int i = 0; i < 9; ++i) s_w[1 + i] = s_head[i];
        // interior breakpoints kpos[1..6] for interval search
        for (int i = 0; i < 6; ++i) tbl[i] = s_ak[4 + i];
        tbl[6] = 0.f; tbl[7] = 0.f;          // pad (keep d_ws fully written)
    }
    __syncthreads();

    // ---- lanes 0..6: symbolic de Boor -> power-basis cubic per interval ----
    // interval L corresponds to knot index k = 3+L, xp in [ak[k], ak[k+1])
    if (L < 7) {
        const int k = 3 + L;
        double d[4][4];                      // d[j] = cubic coeffs (1,x,x^2,x^3)
        for (int j = 0; j < 4; ++j) {
            d[j][0] = (double)s_w[j + k - 3];
            d[j][1] = d[j][2] = d[j][3] = 0.0;
        }
        for (int r = 1; r <= 3; ++r) {
            for (int j = 3; j >= r; --j) {   // descending j: d[j-1] is level r-1
                const double A   = (double)s_ak[j + k - 3];
                const double den = (double)s_ak[j + 1 + k - r] - A;
                const double c1  = 1.0 / den;          // alpha = c0 + c1*x
                const double c0  = -A * c1;
                double diff[4], nd[4];
                for (int i = 0; i < 4; ++i) diff[i] = d[j][i] - d[j - 1][i];
                for (int i = 0; i < 4; ++i) nd[i] = d[j - 1][i] + diff[i] * c0;
                for (int i = 0; i < 3; ++i) nd[i + 1] += diff[i] * c1;  // diff[3]==0 pre-final
                for (int i = 0; i < 4; ++i) d[j][i] = nd[i];
            }
        }
        for (int i = 0; i < 4; ++i) tbl[8 + 4 * L + i] = (float)d[3][i];
    }
}

__device__ __forceinline__ float eval_spline(float xc, const float* s)
{
    float xp = fminf(fmaxf(xc * 0.57735026918962576f, 0.0f), 0.9999f);
    int idx = 0;
#pragma unroll
    for (int j = 0; j < 6; ++j) idx += (xp >= s[j]) ? 1 : 0;
    const float4 cf = *reinterpret_cast<const float4*>(&s[8 + (idx << 2)]);
    return fmaf(fmaf(fmaf(cf.w, xp, cf.z), xp, cf.y), xp, cf.x);
}

__global__ void __launch_bounds__(256)
spline_eval_kernel(const float* __restrict__ x, const float* __restrict__ tbl,
                   float* __restrict__ out, int n)
{
    __shared__ __align__(16) float s_tbl[36];
    const int t = threadIdx.x;
    if (t < 36) s_tbl[t] = tbl[t];
    __syncthreads();

    const int n4     = n >> 2;
    const int gid    = blockIdx.x * blockDim.x + t;
    const int stride = gridDim.x * blockDim.x;

    const float4* __restrict__ x4 = reinterpret_cast<const float4*>(x);
    float4* __restrict__       o4 = reinterpret_cast<float4*>(out);

    for (int i = gid; i < n4; i += stride) {
        const float4 xv = x4[i];
        float4 r;
        r.x = eval_spline(xv.x, s_tbl);
        r.y = eval_spline(xv.y, s_tbl);
        r.z = eval_spline(xv.z, s_tbl);
        r.w = eval_spline(xv.w, s_tbl);
        o4[i] = r;
    }
    // scalar tail (n not divisible by 4)
    const int rem = n & 3;
    if (gid < rem) {
        const int idx = (n4 << 2) + gid;
        out[idx] = eval_spline(x[idx], s_tbl);
    }
}

extern "C" void kernel_launch(void* const* d_in, const int* in_sizes, int n_in,
                              void* d_out, int out_size, void* d_ws, size_t ws_size,
                              hipStream_t stream)
{
    const float* x  = (const float*)d_in[0];
    const float* a  = (const float*)d_in[1];
    const float* W1 = (const float*)d_in[2];
    const float* b1 = (const float*)d_in[3];
    const float* W2 = (const float*)d_in[4];
    const float* b2 = (const float*)d_in[5];
    const float* Ww = (const float*)d_in[6];
    const float* bw = (const float*)d_in[7];
    const float* Wk = (const float*)d_in[8];
    const float* bk = (const float*)d_in[9];
    float* out = (float*)d_out;
    float* tbl = (float*)d_ws;

    mlp_setup_kernel<<<1, 32, 0, stream>>>(a, W1, b1, W2, b2, Ww, bw, Wk, bk, tbl);

    const int n   = out_size;
    const int n4  = n >> 2;
    int threads   = 256;
    int blocks    = (n4 + threads - 1) / threads;
    if (blocks < 1) blocks = 1;
    spline_eval_kernel<<<blocks, threads, 0, stream>>>(x, tbl, out, n);
}